// GAT_55405078119117
// MI455X (gfx1250) — compile-verified
//
#include <hip/hip_runtime.h>
#include <hip/hip_bf16.h>

#define N_NODES   100000
#define N_EDGES   600000
#define E_TOT     700000   // edges + self loops
#define NUM_GRAPHS 128
#define DIM_IN    64
#define DIM_OUT   16
#define HDIM      128

typedef __attribute__((ext_vector_type(2))) float v2f;
typedef __attribute__((ext_vector_type(8))) float v8f;

// ---------------------------------------------------------------------------
// FP32 WMMA GEMM: C[M x (NT*16)] = act(A[M x K] @ W[K x NT*16] + bias)
// One wave (32 threads) per 16-row strip; NT accumulator tiles (A-frag reuse).
// V_WMMA_F32_16X16X4_F32 fragment layouts per CDNA5 ISA 7.12.2:
//   A 16x4 : lanes 0-15 row M=l;  v[0]=K0,v[1]=K1 | lanes 16-31: v[0]=K2,v[1]=K3
//   B 4x16 : row-striped like C/D: v[0]=row(k+2*half), v[1]=row(k+2*half+1), col=l&15
//   D 16x16: reg r, lane l -> row r+8*(l>>4), col l&15
// ---------------------------------------------------------------------------
template <int NT>
__global__ __launch_bounds__(32)
void gemm_wmma_f32(const float* __restrict__ A, const float* __restrict__ W,
                   const float* __restrict__ bias, float* __restrict__ C,
                   int K, float slope) {
  const int N    = NT * 16;
  const int lane = threadIdx.x & 31;
  const int half = lane >> 4;
  const int l16  = lane & 15;
  const int m0   = blockIdx.x * 16;

  v8f acc[NT];
#pragma unroll
  for (int t = 0; t < NT; ++t)
#pragma unroll
    for (int r = 0; r < 8; ++r) acc[t][r] = 0.0f;

  const float* arow = A + (size_t)(m0 + l16) * K;
  for (int k = 0; k < K; k += 4) {
    // A fragment: 8-byte vector load (global_load_b64)
    v2f a = *(const v2f*)(arow + k + 2 * half);
#pragma unroll
    for (int t = 0; t < NT; ++t) {
      const float* wp = W + (size_t)(k + 2 * half) * N + t * 16 + l16;
      v2f b; b.x = wp[0]; b.y = wp[N];
      acc[t] = __builtin_amdgcn_wmma_f32_16x16x4_f32(
          false, a, false, b, (short)0, acc[t], false, false);
    }
  }

#pragma unroll
  for (int t = 0; t < NT; ++t) {
    const int col = t * 16 + l16;
    const float bv = bias[col];
#pragma unroll
    for (int r = 0; r < 8; ++r) {
      const int row = m0 + r + 8 * half;
      float v = acc[t][r] + bv;
      v = (v >= 0.0f) ? v : v * slope;
      C[(size_t)row * N + col] = v;
    }
  }
}

// ---------------------------------------------------------------------------
// Utility / edge-phase kernels
// ---------------------------------------------------------------------------
__global__ void fill_kernel(float* __restrict__ p, float v, int n) {
  int i = blockIdx.x * blockDim.x + threadIdx.x;
  if (i < n) p[i] = v;
}

__device__ __forceinline__ void edge_pair(int e, const int* __restrict__ es,
                                          const int* __restrict__ ed,
                                          int& src, int& dst) {
  if (e < N_EDGES) { src = es[e]; dst = ed[e]; }
  else             { src = dst = e - N_EDGES; }   // self loop
}

__device__ __forceinline__ void atomicMaxF(float* addr, float v) {
  if (v >= 0.0f) atomicMax((int*)addr, __float_as_int(v));
  else           atomicMin((unsigned int*)addr, __float_as_uint(v));
}

// e_ij = att . leaky_relu(xl[src] + xr[dst], 0.2) ; one wave per edge
__global__ void edge_logits_kernel(const float* __restrict__ xl,
                                   const float* __restrict__ xr,
                                   const float* __restrict__ att,
                                   const int* __restrict__ es,
                                   const int* __restrict__ ed,
                                   float* __restrict__ elog) {
  int e = (blockIdx.x * blockDim.x + threadIdx.x) >> 5;
  int lane = threadIdx.x & 31;
  if (e >= E_TOT) return;
  int src, dst; edge_pair(e, es, ed, src, dst);
  float4 l = ((const float4*)(xl + (size_t)src * HDIM))[lane];
  float4 r = ((const float4*)(xr + (size_t)dst * HDIM))[lane];
  float4 a = ((const float4*)att)[lane];
  float s = 0.0f, m;
  m = l.x + r.x; s += a.x * (m >= 0.0f ? m : 0.2f * m);
  m = l.y + r.y; s += a.y * (m >= 0.0f ? m : 0.2f * m);
  m = l.z + r.z; s += a.z * (m >= 0.0f ? m : 0.2f * m);
  m = l.w + r.w; s += a.w * (m >= 0.0f ? m : 0.2f * m);
#pragma unroll
  for (int off = 16; off > 0; off >>= 1) s += __shfl_xor(s, off, 32);
  if (lane == 0) elog[e] = s;
}

__global__ void seg_max_kernel(const float* __restrict__ elog,
                               const int* __restrict__ es,
                               const int* __restrict__ ed,
                               float* __restrict__ emax) {
  int e = blockIdx.x * blockDim.x + threadIdx.x;
  if (e >= E_TOT) return;
  int src, dst; edge_pair(e, es, ed, src, dst);
  atomicMaxF(&emax[dst], elog[e]);
}

__global__ void exp_denom_kernel(const float* __restrict__ elog,
                                 const int* __restrict__ es,
                                 const int* __restrict__ ed,
                                 const float* __restrict__ emax,
                                 float* __restrict__ abuf,
                                 float* __restrict__ denom) {
  int e = blockIdx.x * blockDim.x + threadIdx.x;
  if (e >= E_TOT) return;
  int src, dst; edge_pair(e, es, ed, src, dst);
  float a = expf(elog[e] - emax[dst]);
  abuf[e] = a;
  atomicAdd(&denom[dst], a);
}

// out[dst] += (a/denom[dst]) * xl[src] ; one wave per edge, f32 atomics
__global__ void aggregate_kernel(const float* __restrict__ xl,
                                 const float* __restrict__ abuf,
                                 const float* __restrict__ denom,
                                 const int* __restrict__ es,
                                 const int* __restrict__ ed,
                                 float* __restrict__ out) {
  int e = (blockIdx.x * blockDim.x + threadIdx.x) >> 5;
  int lane = threadIdx.x & 31;
  if (e >= E_TOT) return;
  int src, dst; edge_pair(e, es, ed, src, dst);
  float alpha = abuf[e] / denom[dst];
  float4 v = ((const float4*)(xl + (size_t)src * HDIM))[lane];
  float* o = out + (size_t)dst * HDIM + lane * 4;
  atomicAdd(o + 0, alpha * v.x);
  atomicAdd(o + 1, alpha * v.y);
  atomicAdd(o + 2, alpha * v.z);
  atomicAdd(o + 3, alpha * v.w);
}

__global__ void bias_act_kernel(float* __restrict__ h, const float* __restrict__ bias,
                                int n, float slope) {
  int i = blockIdx.x * blockDim.x + threadIdx.x;
  if (i >= n) return;
  float v = h[i] + bias[i & (HDIM - 1)];
  h[i] = (v >= 0.0f) ? v : v * slope;
}

// global mean pool: one wave per node
__global__ void pool_kernel(const float* __restrict__ h, const int* __restrict__ batch,
                            float* __restrict__ sums, float* __restrict__ counts) {
  int node = (blockIdx.x * blockDim.x + threadIdx.x) >> 5;
  int lane = threadIdx.x & 31;
  if (node >= N_NODES) return;
  int g = batch[node];
  float4 v = ((const float4*)(h + (size_t)node * HDIM))[lane];
  float* s = sums + (size_t)g * HDIM + lane * 4;
  atomicAdd(s + 0, v.x);
  atomicAdd(s + 1, v.y);
  atomicAdd(s + 2, v.z);
  atomicAdd(s + 3, v.w);
  if (lane == 0) atomicAdd(&counts[g], 1.0f);
}

__global__ void mean_kernel(const float* __restrict__ sums, const float* __restrict__ counts,
                            float* __restrict__ hg) {
  int i = blockIdx.x * blockDim.x + threadIdx.x;
  if (i >= NUM_GRAPHS * HDIM) return;
  hg[i] = sums[i] / fmaxf(counts[i >> 7], 1.0f);
}

// ---------------------------------------------------------------------------
extern "C" void kernel_launch(void* const* d_in, const int* in_sizes, int n_in,
                              void* d_out, int out_size, void* d_ws, size_t ws_size,
                              hipStream_t stream) {
  (void)in_sizes; (void)n_in; (void)out_size; (void)ws_size;

  const float* x      = (const float*)d_in[0];
  const int*   eidx   = (const int*)d_in[1];     // [2, N_EDGES]
  const int*   batch  = (const int*)d_in[2];
  const float* W_nfc  = (const float*)d_in[3];
  const float* b_nfc  = (const float*)d_in[4];
  const float* Wl[2]  = {(const float*)d_in[5],  (const float*)d_in[11]};
  const float* bl[2]  = {(const float*)d_in[6],  (const float*)d_in[12]};
  const float* Wr[2]  = {(const float*)d_in[7],  (const float*)d_in[13]};
  const float* br[2]  = {(const float*)d_in[8],  (const float*)d_in[14]};
  const float* att[2] = {(const float*)d_in[9],  (const float*)d_in[15]};
  const float* bias[2]= {(const float*)d_in[10], (const float*)d_in[16]};
  const float* W_fc1  = (const float*)d_in[17];
  const float* b_fc1  = (const float*)d_in[18];
  const float* W_fc2  = (const float*)d_in[19];
  const float* b_fc2  = (const float*)d_in[20];
  float* out = (float*)d_out;

  const int* e_src = eidx;
  const int* e_dst = eidx + N_EDGES;

  // workspace layout (floats); total ~160 MB
  const size_t NH = (size_t)N_NODES * HDIM;
  float* ws      = (float*)d_ws;
  float* buf_h   = ws;                    // node features / aggregation target
  float* buf_xl  = buf_h  + NH;
  float* buf_xr  = buf_xl + NH;
  float* e_log   = buf_xr + NH;           // E_TOT
  float* a_buf   = e_log  + E_TOT;        // E_TOT
  float* emax    = a_buf  + E_TOT;        // N_NODES
  float* denom   = emax   + N_NODES;      // N_NODES
  float* sums    = denom  + N_NODES;      // NUM_GRAPHS*HDIM
  float* counts  = sums   + NUM_GRAPHS * HDIM;  // NUM_GRAPHS
  float* hg      = counts + NUM_GRAPHS;         // NUM_GRAPHS*HDIM
  float* h1      = hg     + NUM_GRAPHS * HDIM;  // NUM_GRAPHS*32

  const int TPB = 256;
  const int g_nh    = (int)((NH + TPB - 1) / TPB);
  const int g_nodes = (N_NODES + TPB - 1) / TPB;
  const int g_edge1 = (E_TOT + TPB - 1) / TPB;
  const int g_edgeW = (int)(((size_t)E_TOT * 32 + TPB - 1) / TPB);
  const int g_nodeW = (int)(((size_t)N_NODES * 32 + TPB - 1) / TPB);

  // node feature MLP: hx = leaky_relu(x @ W_nfc + b, 0.01)
  gemm_wmma_f32<8><<<N_NODES / 16, 32, 0, stream>>>(x, W_nfc, b_nfc, buf_h, DIM_IN, 0.01f);

  for (int L = 0; L < 2; ++L) {
    gemm_wmma_f32<8><<<N_NODES / 16, 32, 0, stream>>>(buf_h, Wl[L], bl[L], buf_xl, HDIM, 1.0f);
    gemm_wmma_f32<8><<<N_NODES / 16, 32, 0, stream>>>(buf_h, Wr[L], br[L], buf_xr, HDIM, 1.0f);
    fill_kernel<<<g_nh,    TPB, 0, stream>>>(buf_h, 0.0f, (int)NH);
    fill_kernel<<<g_nodes, TPB, 0, stream>>>(emax, -__builtin_huge_valf(), N_NODES);
    fill_kernel<<<g_nodes, TPB, 0, stream>>>(denom, 0.0f, N_NODES);
    edge_logits_kernel<<<g_edgeW, TPB, 0, stream>>>(buf_xl, buf_xr, att[L], e_src, e_dst, e_log);
    seg_max_kernel   <<<g_edge1, TPB, 0, stream>>>(e_log, e_src, e_dst, emax);
    exp_denom_kernel <<<g_edge1, TPB, 0, stream>>>(e_log, e_src, e_dst, emax, a_buf, denom);
    aggregate_kernel <<<g_edgeW, TPB, 0, stream>>>(buf_xl, a_buf, denom, e_src, e_dst, buf_h);
    bias_act_kernel  <<<g_nh,    TPB, 0, stream>>>(buf_h, bias[L], (int)NH, 0.01f);
  }

  // global mean pool + head
  fill_kernel<<<(NUM_GRAPHS * HDIM + TPB - 1) / TPB, TPB, 0, stream>>>(sums, 0.0f, NUM_GRAPHS * HDIM);
  fill_kernel<<<1, NUM_GRAPHS, 0, stream>>>(counts, 0.0f, NUM_GRAPHS);
  pool_kernel<<<g_nodeW, TPB, 0, stream>>>(buf_h, batch, sums, counts);
  mean_kernel<<<(NUM_GRAPHS * HDIM + TPB - 1) / TPB, TPB, 0, stream>>>(sums, counts, hg);

  gemm_wmma_f32<2><<<NUM_GRAPHS / 16, 32, 0, stream>>>(hg, W_fc1, b_fc1, h1, HDIM, 0.01f);
  gemm_wmma_f32<1><<<NUM_GRAPHS / 16, 32, 0, stream>>>(h1, W_fc2, b_fc2, out, 32, 1.0f);
}